// Mamba_Momentum_Direct_36859409334279
// MI455X (gfx1250) — compile-verified
//
#include <hip/hip_runtime.h>
#include <math.h>

#define D_MODEL 128
#define ED      256
#define NSTATE  64
#define DT_RANK 8
#define DCONV   4
#define DBL_W   (DT_RANK + 2*NSTATE)   // 136
#define NLAYER  2
#define BATCH   64
#define LSEQ    2048
#define BLROWS  (BATCH*LSEQ)           // 131072
#define LCHUNK  16

typedef float v2f __attribute__((ext_vector_type(2)));
typedef float v8f __attribute__((ext_vector_type(8)));

__device__ __forceinline__ float sigmoidf_(float x){ return 1.f/(1.f+__expf(-x)); }

#define WMMA_F32(a, b, c) \
    __builtin_amdgcn_wmma_f32_16x16x4_f32(false, (a), false, (b), (short)0, (c), false, false)

// ---------------------------------------------------------------------------
// Front-end: conv1d(pad=1, Cin=6, Cout=128, k=3) + BN + ReLU, write (B,L,128)
// ---------------------------------------------------------------------------
__global__ void __launch_bounds__(D_MODEL)
front_conv(const float* __restrict__ x, const float* __restrict__ cw,
           const float* __restrict__ g, const float* __restrict__ bta,
           float* __restrict__ h)
{
    __shared__ float sx[18];
    int bl = blockIdx.x;
    int b = bl / LSEQ, l = bl % LSEQ;
    int d = threadIdx.x;
    if (d < 18) {
        int c = d / 3, j = d % 3, li = l - 1 + j;
        sx[d] = (li >= 0 && li < LSEQ) ? x[((size_t)b*6 + c)*LSEQ + li] : 0.f;
    }
    __syncthreads();
    const float* w = cw + d*18;
    float s = 0.f;
#pragma unroll
    for (int t = 0; t < 18; ++t) s += w[t]*sx[t];
    s = g[d]*s + bta[d];
    h[(size_t)bl*D_MODEL + d] = fmaxf(s, 0.f);
}

// ---------------------------------------------------------------------------
// RMSNorm over last dim (128), one row per 128-thread block
// ---------------------------------------------------------------------------
__global__ void __launch_bounds__(D_MODEL)
rmsnorm_k(const float* __restrict__ hin, const float* __restrict__ w,
          float* __restrict__ outp)
{
    __shared__ float red[4];
    int bl = blockIdx.x, d = threadIdx.x;
    float v = hin[(size_t)bl*D_MODEL + d];
    float ss = v*v;
    for (int off = 16; off; off >>= 1) ss += __shfl_xor(ss, off, 32);
    if ((d & 31) == 0) red[d >> 5] = ss;
    __syncthreads();
    float tot = red[0] + red[1] + red[2] + red[3];
    float inv = 1.f/(sqrtf(tot)*0.08838834764831845f + 1e-6f);  // 128^-0.5
    outp[(size_t)bl*D_MODEL + d] = w[d]*v*inv;
}

// ---------------------------------------------------------------------------
// WMMA f32 GEMM: C[M,No] = A[M,K] @ W[Nw,K]^T (+ res), ldc stride.
// 4 waves/block; wave = 32Mx64N tile (2 A frags x 4 B frags = 8 wmma/k-step).
// Per k-step: 6 x global_load_b64 feed 8 x v_wmma_f32_16x16x4_f32.
// Full-tile blocks (uniform branch on blockIdx.x) take the mask-free path;
// only the ragged last N-tile pays the multiplicative-mask VALU cost.
// A frag: lane m=lane&15, VGPR v <-> k = k0 + v + 2*(lane>>4)
// B frag: lane n=lane&15, VGPR v <-> k = k0 + v + 2*(lane>>4)
// ---------------------------------------------------------------------------
__global__ void __launch_bounds__(128)
wmma_gemm_f32(const float* __restrict__ A, const float* __restrict__ W,
              const float* res, float* C,
              int K, int Nw, int No, int ldc)
{
    int lane  = threadIdx.x & 31;
    int wid   = threadIdx.x >> 5;
    int m0    = (blockIdx.y*4 + wid)*32;
    int n0    = blockIdx.x*64;
    int half  = lane >> 4;
    int idx16 = lane & 15;

    const float* ap0 = A + (size_t)(m0 + idx16)*K + 2*half;
    const float* ap1 = ap0 + (size_t)16*K;

    v8f acc[8] = {};

    bool fullN = (n0 + 64 <= Nw);   // uniform across block

    if (fullN) {
        const float* wp0 = W + (size_t)(n0 + 0*16 + idx16)*K + 2*half;
        const float* wp1 = W + (size_t)(n0 + 1*16 + idx16)*K + 2*half;
        const float* wp2 = W + (size_t)(n0 + 2*16 + idx16)*K + 2*half;
        const float* wp3 = W + (size_t)(n0 + 3*16 + idx16)*K + 2*half;
        for (int k = 0; k < K; k += 4) {
            v2f a0 = *(const v2f*)(ap0 + k);
            v2f a1 = *(const v2f*)(ap1 + k);
            v2f b0 = *(const v2f*)(wp0 + k);
            v2f b1 = *(const v2f*)(wp1 + k);
            v2f b2 = *(const v2f*)(wp2 + k);
            v2f b3 = *(const v2f*)(wp3 + k);
            acc[0] = WMMA_F32(a0, b0, acc[0]);
            acc[1] = WMMA_F32(a1, b0, acc[1]);
            acc[2] = WMMA_F32(a0, b1, acc[2]);
            acc[3] = WMMA_F32(a1, b1, acc[3]);
            acc[4] = WMMA_F32(a0, b2, acc[4]);
            acc[5] = WMMA_F32(a1, b2, acc[5]);
            acc[6] = WMMA_F32(a0, b3, acc[6]);
            acc[7] = WMMA_F32(a1, b3, acc[7]);
        }
    } else {
        const float* wp[4]; float msk[4];
#pragma unroll
        for (int j = 0; j < 4; ++j) {
            int n = n0 + j*16 + idx16;
            msk[j] = (n < Nw) ? 1.f : 0.f;
            wp[j]  = W + (size_t)((n < Nw) ? n : 0)*K + 2*half;
        }
        for (int k = 0; k < K; k += 4) {
            v2f a0 = *(const v2f*)(ap0 + k);
            v2f a1 = *(const v2f*)(ap1 + k);
            v2f b0 = *(const v2f*)(wp[0] + k) * msk[0];
            v2f b1 = *(const v2f*)(wp[1] + k) * msk[1];
            v2f b2 = *(const v2f*)(wp[2] + k) * msk[2];
            v2f b3 = *(const v2f*)(wp[3] + k) * msk[3];
            acc[0] = WMMA_F32(a0, b0, acc[0]);
            acc[1] = WMMA_F32(a1, b0, acc[1]);
            acc[2] = WMMA_F32(a0, b1, acc[2]);
            acc[3] = WMMA_F32(a1, b1, acc[3]);
            acc[4] = WMMA_F32(a0, b2, acc[4]);
            acc[5] = WMMA_F32(a1, b2, acc[5]);
            acc[6] = WMMA_F32(a0, b3, acc[6]);
            acc[7] = WMMA_F32(a1, b3, acc[7]);
        }
    }

#pragma unroll
    for (int j = 0; j < 4; ++j) {
        int n = n0 + j*16 + idx16;
        if (n < No) {
#pragma unroll
            for (int i = 0; i < 2; ++i) {
#pragma unroll
                for (int r = 0; r < 8; ++r) {
                    int m = m0 + i*16 + r + 8*half;  // C/D layout: row = vgpr + 8*(lane>=16)
                    float v = acc[j*2 + i][r];
                    if (res) v += res[(size_t)m*ldc + n];
                    C[(size_t)m*ldc + n] = v;
                }
            }
        }
    }
}

// ---------------------------------------------------------------------------
// Depthwise causal conv (k=4) over L on xin part of xz (cols 0..255) + SiLU
// ---------------------------------------------------------------------------
__global__ void __launch_bounds__(ED)
dwconv_silu(const float* __restrict__ xz, const float* __restrict__ cw,
            const float* __restrict__ cb, float* __restrict__ xin)
{
    int b = blockIdx.x, lc = blockIdx.y, e = threadIdx.x;
    int l0 = lc*128;
    const float* base = xz + (size_t)b*LSEQ*(2*ED) + e;
    float k0 = cw[e*DCONV+0], k1 = cw[e*DCONV+1], k2 = cw[e*DCONV+2], k3 = cw[e*DCONV+3];
    float bias = cb[e];
    float w0 = (l0-3 >= 0) ? base[(size_t)(l0-3)*(2*ED)] : 0.f;
    float w1 = (l0-2 >= 0) ? base[(size_t)(l0-2)*(2*ED)] : 0.f;
    float w2 = (l0-1 >= 0) ? base[(size_t)(l0-1)*(2*ED)] : 0.f;
    for (int l = l0; l < l0 + 128; ++l) {
        float w3 = base[(size_t)l*(2*ED)];
        float s = k0*w0 + k1*w1 + k2*w2 + k3*w3 + bias;
        xin[((size_t)b*LSEQ + l)*ED + e] = s * sigmoidf_(s);
        w0 = w1; w1 = w2; w2 = w3;
    }
}

// ---------------------------------------------------------------------------
// delta = softplus(dr @ dt_w.T + dt_b), dr = dbl[:, 0:8]
// ---------------------------------------------------------------------------
__global__ void __launch_bounds__(ED)
delta_softplus(const float* __restrict__ dbl, const float* __restrict__ dtw,
               const float* __restrict__ dtb, float* __restrict__ dly)
{
    int e = threadIdx.x;
    size_t row = blockIdx.x;
    const float* dr = dbl + row*DBL_W;
    float s = dtb[e];
#pragma unroll
    for (int r = 0; r < DT_RANK; ++r) s += dr[r]*dtw[e*DT_RANK + r];
    dly[row*ED + e] = (s > 20.f) ? s : log1pf(__expf(s));
}

// ---------------------------------------------------------------------------
// Selective scan with momentum. Block = (batch b, 8 e-channels), wave per
// channel, 2 states per lane. B_t/C_t staged in LDS once per step and shared
// by 8 waves. y written in-place over delta (same element, same wave).
// ---------------------------------------------------------------------------
__global__ void __launch_bounds__(256)
scan_kernel(const float* __restrict__ dbl, const float* dly /* delta in */,
            const float* __restrict__ xin, const float* __restrict__ A_log,
            float* yout /* may alias dly */)
{
    __shared__ float sBC[128];     // [0..63]=B_t, [64..127]=C_t
    int b    = blockIdx.x;
    int wid  = threadIdx.x >> 5, lane = threadIdx.x & 31;
    int e    = blockIdx.y*8 + wid;
    float A0 = -__expf(A_log[e*NSTATE + lane]);
    float A1 = -__expf(A_log[e*NSTATE + lane + 32]);
    float h0 = 0.f, h1 = 0.f, p0 = 0.f, p1 = 0.f;

    for (int l = 0; l < LSEQ; ++l) {
        size_t row = (size_t)b*LSEQ + l;
        if (threadIdx.x < 128)
            sBC[threadIdx.x] = dbl[row*DBL_W + DT_RANK + threadIdx.x];
        __syncthreads();

        float d  = dly[row*ED + e];
        float x  = xin[row*ED + e];
        float dx = d*x;
        p0 = 0.8f*p0 + dx*sBC[lane];
        p1 = 0.8f*p1 + dx*sBC[lane + 32];
        h0 = __expf(d*A0)*h0 + p0;                 // ALPHA = 1
        h1 = __expf(d*A1)*h1 + p1;
        float part = h0*sBC[64 + lane] + h1*sBC[96 + lane];
        for (int off = 16; off; off >>= 1) part += __shfl_xor(part, off, 32);
        if (lane == 0) yout[row*ED + e] = part;
        __syncthreads();
    }
}

// ---------------------------------------------------------------------------
// yg = (y + xin*Dp) * silu(z);  z = xz[:, 256:512]; written in-place to xin
// ---------------------------------------------------------------------------
__global__ void __launch_bounds__(ED)
gate_kernel(const float* y, const float* __restrict__ xz,
            const float* __restrict__ Dp, float* xin)
{
    int e = threadIdx.x;
    size_t row = blockIdx.x;
    float z = xz[row*(2*ED) + ED + e];
    float v = y[row*ED + e] + xin[row*ED + e]*Dp[e];
    xin[row*ED + e] = v * (z * sigmoidf_(z));
}

// ---------------------------------------------------------------------------
// Head: row-wise 1/(norm+eps), chunked mean accumulation, classifier
// ---------------------------------------------------------------------------
__global__ void __launch_bounds__(D_MODEL)
row_inv(const float* __restrict__ h, float* __restrict__ inv)
{
    __shared__ float red[4];
    int bl = blockIdx.x, d = threadIdx.x;
    float v = h[(size_t)bl*D_MODEL + d];
    float ss = v*v;
    for (int off = 16; off; off >>= 1) ss += __shfl_xor(ss, off, 32);
    if ((d & 31) == 0) red[d >> 5] = ss;
    __syncthreads();
    if (d == 0) {
        float tot = red[0] + red[1] + red[2] + red[3];
        inv[bl] = 1.f/(sqrtf(tot)*0.08838834764831845f + 1e-6f);
    }
}

__global__ void __launch_bounds__(D_MODEL)
mean_partial(const float* __restrict__ h, const float* __restrict__ inv,
             float* __restrict__ part)
{
    int b = blockIdx.x, ch = blockIdx.y, d = threadIdx.x;
    const int lpc = LSEQ/LCHUNK;
    float s = 0.f;
    for (int l = ch*lpc; l < (ch+1)*lpc; ++l) {
        size_t row = (size_t)b*LSEQ + l;
        s += h[row*D_MODEL + d]*inv[row];
    }
    part[((size_t)b*LCHUNK + ch)*D_MODEL + d] = s;
}

__global__ void __launch_bounds__(32)
head_kernel(const float* __restrict__ part, const float* __restrict__ fw,
            const float* __restrict__ cw, const float* __restrict__ cb,
            float* __restrict__ outp)
{
    int b = blockIdx.x, c = threadIdx.x;
    float s = cb[c];
    for (int d = 0; d < D_MODEL; ++d) {
        float t = 0.f;
#pragma unroll
        for (int ch = 0; ch < LCHUNK; ++ch)
            t += part[((size_t)b*LCHUNK + ch)*D_MODEL + d];
        s += t * (fw[d]*(1.f/(float)LSEQ)) * cw[c*D_MODEL + d];
    }
    outp[b*32 + c] = s;
}

// ---------------------------------------------------------------------------
extern "C" void kernel_launch(void* const* d_in, const int* in_sizes, int n_in,
                              void* d_out, int out_size, void* d_ws, size_t ws_size,
                              hipStream_t stream)
{
    const float* x     = (const float*)d_in[0];
    const float* convw = (const float*)d_in[1];
    const float* bng   = (const float*)d_in[2];
    const float* bnb   = (const float*)d_in[3];
    const float* inw   = (const float*)d_in[4];
    const float* cw    = (const float*)d_in[5];
    const float* cb    = (const float*)d_in[6];
    const float* xpw   = (const float*)d_in[7];
    const float* dtw   = (const float*)d_in[8];
    const float* dtb   = (const float*)d_in[9];
    const float* alog  = (const float*)d_in[10];
    const float* Dp    = (const float*)d_in[11];
    const float* outw  = (const float*)d_in[12];
    const float* lnw   = (const float*)d_in[13];
    const float* fnw   = (const float*)d_in[14];
    const float* clsw  = (const float*)d_in[15];
    const float* clsb  = (const float*)d_in[16];
    float* outp = (float*)d_out;

    char* ws = (char*)d_ws;
    size_t off = 0;
    auto alloc = [&](size_t bytes) -> float* {
        float* p = (float*)(ws + off);
        off += (bytes + 255) & ~(size_t)255;
        return p;
    };
    float* h_    = alloc((size_t)BLROWS*D_MODEL*4);
    float* u_    = alloc((size_t)BLROWS*D_MODEL*4);
    float* xz_   = alloc((size_t)BLROWS*2*ED*4);
    float* xin_  = alloc((size_t)BLROWS*ED*4);
    float* dbl_  = alloc((size_t)BLROWS*DBL_W*4);
    float* dly_  = alloc((size_t)BLROWS*ED*4);
    float* inv_  = alloc((size_t)BLROWS*4);
    float* part_ = alloc((size_t)BATCH*LCHUNK*D_MODEL*4);

    front_conv<<<BLROWS, D_MODEL, 0, stream>>>(x, convw, bng, bnb, h_);

    for (int i = 0; i < NLAYER; ++i) {
        rmsnorm_k<<<BLROWS, D_MODEL, 0, stream>>>(h_, lnw + i*D_MODEL, u_);
        // xz = u @ in_w[i].T : M=131072, K=128, N=512
        wmma_gemm_f32<<<dim3((2*ED)/64, BLROWS/128), 128, 0, stream>>>(
            u_, inw + (size_t)i*2*ED*D_MODEL, nullptr, xz_, D_MODEL, 2*ED, 2*ED, 2*ED);
        dwconv_silu<<<dim3(BATCH, LSEQ/128), ED, 0, stream>>>(
            xz_, cw + (size_t)i*ED*DCONV, cb + i*ED, xin_);
        // dbl = xin @ xp_w[i].T : M=131072, K=256, N=136 (ragged last tile)
        wmma_gemm_f32<<<dim3((DBL_W + 63)/64, BLROWS/128), 128, 0, stream>>>(
            xin_, xpw + (size_t)i*DBL_W*ED, nullptr, dbl_, ED, DBL_W, DBL_W, DBL_W);
        delta_softplus<<<BLROWS, ED, 0, stream>>>(
            dbl_, dtw + (size_t)i*ED*DT_RANK, dtb + i*ED, dly_);
        scan_kernel<<<dim3(BATCH, ED/8), 256, 0, stream>>>(
            dbl_, dly_, xin_, alog + (size_t)i*ED*NSTATE, dly_);
        gate_kernel<<<BLROWS, ED, 0, stream>>>(dly_, xz_, Dp + i*ED, xin_);
        // h += yg @ out_w[i].T : M=131072, K=256, N=128 (residual fused)
        wmma_gemm_f32<<<dim3(D_MODEL/64, BLROWS/128), 128, 0, stream>>>(
            xin_, outw + (size_t)i*D_MODEL*ED, h_, h_, ED, D_MODEL, D_MODEL, D_MODEL);
    }

    row_inv<<<BLROWS, D_MODEL, 0, stream>>>(h_, inv_);
    mean_partial<<<dim3(BATCH, LCHUNK), D_MODEL, 0, stream>>>(h_, inv_, part_);
    head_kernel<<<BATCH, 32, 0, stream>>>(part_, fnw, clsw, clsb, outp);
}